// OrthogonalLipschitzLinear_77764677862195
// MI455X (gfx1250) — compile-verified
//
#include <hip/hip_runtime.h>
#include <math.h>

// Problem sizes (fixed by the reference)
#define IN_F   1024
#define OUT_F  1024
#define M_ROWS 16384

typedef float v2f __attribute__((ext_vector_type(2)));
typedef float v8f __attribute__((ext_vector_type(8)));

// ---------------------------------------------------------------------------
// Kernel 1: normalize the Householder vectors:  vn[i] = v[i] / (||v[i]|| + 1e-8)
// One block per reflector row; 256 threads each cover 4 floats.
// ---------------------------------------------------------------------------
__global__ __launch_bounds__(256) void olll_normalize(const float* __restrict__ hv,
                                                      float* __restrict__ vn) {
  const int row  = blockIdx.x;
  const int tid  = threadIdx.x;
  const int lane = tid & 31;
  const int wid  = tid >> 5;

  const float4 v = ((const float4*)(hv + (size_t)row * IN_F))[tid];
  float ss = v.x * v.x + v.y * v.y + v.z * v.z + v.w * v.w;
#pragma unroll
  for (int off = 16; off > 0; off >>= 1) ss += __shfl_xor(ss, off, 32);

  __shared__ float red[8];
  if (lane == 0) red[wid] = ss;
  __syncthreads();
  float tot = 0.0f;
#pragma unroll
  for (int i = 0; i < 8; ++i) tot += red[i];

  const float scale = 1.0f / (sqrtf(tot) + 1e-8f);
  float4 o;
  o.x = v.x * scale; o.y = v.y * scale; o.z = v.z * scale; o.w = v.w * scale;
  ((float4*)(vn + (size_t)row * IN_F))[tid] = o;
}

// ---------------------------------------------------------------------------
// Kernel 2: build W = H1*H2*...*H1024  (rows of W are independent!)
// Row r starts as e_r and evolves w <- w - 2*(w.v_i)*v_i for i = 0..1023.
// One wave (32 lanes) owns one row kept entirely in registers (8 float4/lane).
// 8 waves per block -> 128 blocks cover all 1024 rows.
// ---------------------------------------------------------------------------
__global__ __launch_bounds__(256) void olll_buildw(const float* __restrict__ vn,
                                                   float* __restrict__ W) {
  const int lane = threadIdx.x & 31;
  const int wid  = threadIdx.x >> 5;
  const int r    = blockIdx.x * 8 + wid;

  // chunk j covers columns 128*j + 4*lane + {0..3}
  float4 w4[8];
#pragma unroll
  for (int j = 0; j < 8; ++j) {
    const int c0 = 128 * j + 4 * lane;
    w4[j].x = (c0 + 0 == r) ? 1.0f : 0.0f;
    w4[j].y = (c0 + 1 == r) ? 1.0f : 0.0f;
    w4[j].z = (c0 + 2 == r) ? 1.0f : 0.0f;
    w4[j].w = (c0 + 3 == r) ? 1.0f : 0.0f;
  }

  for (int i = 0; i < IN_F; ++i) {
    const float4* vp = (const float4*)(vn + (size_t)i * IN_F);
    float4 v4[8];
    float  d = 0.0f;
#pragma unroll
    for (int j = 0; j < 8; ++j) {
      v4[j] = vp[32 * j + lane];  // coalesced 512B per j across the wave
      d += w4[j].x * v4[j].x + w4[j].y * v4[j].y +
           w4[j].z * v4[j].z + w4[j].w * v4[j].w;
    }
#pragma unroll
    for (int off = 16; off > 0; off >>= 1) d += __shfl_xor(d, off, 32);
    const float t = 2.0f * d;
#pragma unroll
    for (int j = 0; j < 8; ++j) {
      w4[j].x -= t * v4[j].x;
      w4[j].y -= t * v4[j].y;
      w4[j].z -= t * v4[j].z;
      w4[j].w -= t * v4[j].w;
    }
  }

  float4* wp = (float4*)(W + (size_t)r * IN_F);
#pragma unroll
  for (int j = 0; j < 8; ++j) wp[32 * j + lane] = w4[j];
}

// ---------------------------------------------------------------------------
// Kernel 3: y = x @ W^T + bias via V_WMMA_F32_16X16X4_F32.
// Block tile 128x128, K-step 32, DOUBLE-BUFFERED LDS (2 x 36KB) with padded
// row stride 36 (bank-conflict-free b64 frag loads, 16B-aligned stores).
// 8 waves in a 2x4 grid; each wave owns a 64x32 patch = v8f acc[4][2].
// Per iteration: issue global loads for tile t+1, compute tile t (64 WMMAs),
// store staged regs into the other buffer, single barrier.
// ---------------------------------------------------------------------------
#define BM  128
#define BN  128
#define BK  32
#define LDT 36  // padded LDS row stride (floats)

__global__ __launch_bounds__(256) void olll_gemm(const float* __restrict__ x,
                                                 const float* __restrict__ W,
                                                 const float* __restrict__ bias,
                                                 float* __restrict__ out) {
  __shared__ __align__(16) float sA[2][BM * LDT];
  __shared__ __align__(16) float sB[2][BN * LDT];

  const int tid  = threadIdx.x;
  const int lane = tid & 31;
  const int wid  = tid >> 5;
  const int hi   = lane >> 4;   // 0: lanes 0-15, 1: lanes 16-31
  const int lr   = lane & 15;

  const int m0 = blockIdx.y * BM;
  const int n0 = blockIdx.x * BN;
  const int wm = (wid >> 2) * 64;  // 0 or 64
  const int wn = (wid & 3) * 32;   // 0,32,64,96

  v8f acc[4][2];
#pragma unroll
  for (int m = 0; m < 4; ++m)
#pragma unroll
    for (int n = 0; n < 2; ++n) acc[m][n] = (v8f)0.0f;

  // cooperative staging map: thread -> (row = tid/8 + 32q, col4 = (tid&7)*4)
  const int lrow = tid >> 3;
  const int lcol = (tid & 7) * 4;

  const int NT = IN_F / BK;  // 32 K-tiles
  float4 ra[4], rb[4];

  // ---- prologue: stage tile 0 into buffer 0 ----
#pragma unroll
  for (int q = 0; q < 4; ++q) {
    const int row = lrow + 32 * q;
    ra[q] = *(const float4*)&x[(size_t)(m0 + row) * IN_F + lcol];
    rb[q] = *(const float4*)&W[(size_t)(n0 + row) * IN_F + lcol];
  }
#pragma unroll
  for (int q = 0; q < 4; ++q) {
    const int row = lrow + 32 * q;
    *(float4*)&sA[0][row * LDT + lcol] = ra[q];
    *(float4*)&sB[0][row * LDT + lcol] = rb[q];
  }
  __syncthreads();

  for (int t = 0; t < NT; ++t) {
    const int  cur  = t & 1;
    const bool more = (t + 1 < NT);

    // issue global loads for tile t+1 (overlaps the WMMA stream below)
    if (more) {
      const int kt = (t + 1) * BK;
#pragma unroll
      for (int q = 0; q < 4; ++q) {
        const int row = lrow + 32 * q;
        ra[q] = *(const float4*)&x[(size_t)(m0 + row) * IN_F + kt + lcol];
        rb[q] = *(const float4*)&W[(size_t)(n0 + row) * IN_F + kt + lcol];
      }
      if (t + 2 < NT) {  // uniform; emits global_prefetch_b8 for tile t+2
        __builtin_prefetch(&x[(size_t)(m0 + lrow) * IN_F + kt + BK + lcol], 0, 1);
        __builtin_prefetch(&W[(size_t)(n0 + lrow) * IN_F + kt + BK + lcol], 0, 1);
      }
    }

    // ---- compute tile t from buffer `cur` ----
#pragma unroll
    for (int kk = 0; kk < BK; kk += 4) {
      // A 16x4 frag: lanes 0-15 hold K=kk,kk+1; lanes 16-31 hold K=kk+2,kk+3
      v2f a[4], b[2];
#pragma unroll
      for (int m = 0; m < 4; ++m)
        a[m] = *(const v2f*)&sA[cur][(wm + m * 16 + lr) * LDT + kk + 2 * hi];
      // B 4x16 frag from W rows (B[k][n] = W[n][k]) — mirrored layout
#pragma unroll
      for (int n = 0; n < 2; ++n)
        b[n] = *(const v2f*)&sB[cur][(wn + n * 16 + lr) * LDT + kk + 2 * hi];
#pragma unroll
      for (int m = 0; m < 4; ++m)
#pragma unroll
        for (int n = 0; n < 2; ++n)
          acc[m][n] = __builtin_amdgcn_wmma_f32_16x16x4_f32(
              false, a[m], false, b[n], (short)0, acc[m][n], false, false);
    }

    // ---- stage tile t+1 into the other buffer; one barrier per iteration ----
    if (more) {
#pragma unroll
      for (int q = 0; q < 4; ++q) {
        const int row = lrow + 32 * q;
        *(float4*)&sA[1 - cur][row * LDT + lcol] = ra[q];
        *(float4*)&sB[1 - cur][row * LDT + lcol] = rb[q];
      }
    }
    __syncthreads();
  }

  // Epilogue: C layout — VGPR j holds rows (j, j+8); lanes give N = lr.
  float bv[2];
#pragma unroll
  for (int n = 0; n < 2; ++n) bv[n] = bias[n0 + wn + n * 16 + lr];
#pragma unroll
  for (int m = 0; m < 4; ++m)
#pragma unroll
    for (int n = 0; n < 2; ++n)
#pragma unroll
      for (int j = 0; j < 8; ++j) {
        const int row = m0 + wm + m * 16 + j + 8 * hi;
        const int col = n0 + wn + n * 16 + lr;
        out[(size_t)row * OUT_F + col] = acc[m][n][j] + bv[n];
      }
}

// ---------------------------------------------------------------------------
extern "C" void kernel_launch(void* const* d_in, const int* in_sizes, int n_in,
                              void* d_out, int out_size, void* d_ws, size_t ws_size,
                              hipStream_t stream) {
  const float* x    = (const float*)d_in[0];  // 16384 x 1024
  const float* hv   = (const float*)d_in[1];  // 1024 x 1024
  const float* bias = (const float*)d_in[2];  // 1024
  float* out = (float*)d_out;                 // 16384 x 1024

  float* vn = (float*)d_ws;                   // 4 MB: normalized reflectors
  float* W  = vn + (size_t)OUT_F * IN_F;      // 4 MB: weight matrix

  olll_normalize<<<OUT_F, 256, 0, stream>>>(hv, vn);
  olll_buildw<<<OUT_F / 8, 256, 0, stream>>>(vn, W);
  olll_gemm<<<dim3(OUT_F / BN, M_ROWS / BM), 256, 0, stream>>>(x, W, bias, out);
}